// OSC_PO_4217657884728
// MI455X (gfx1250) — compile-verified
//
#include <hip/hip_runtime.h>
#include <hip/hip_bf16.h>

typedef __attribute__((ext_vector_type(2))) float v2f;
typedef __attribute__((ext_vector_type(8))) float v8f;
typedef __attribute__((ext_vector_type(4))) unsigned int v4u;
typedef __attribute__((ext_vector_type(8))) int v8i;
typedef __attribute__((ext_vector_type(4))) int v4i;

#define NDIM 512
#define TSTEPS 128
#define KT 32                                                 // K-slab staged via TDM
static constexpr size_t MATF = (size_t)NDIM * NDIM;          // 262144 floats per 512x512
// workspace layout (float offsets)
static constexpr size_t OFF_CA0 = 0;
static constexpr size_t OFF_CA1 = OFF_CA0 + MATF;
static constexpr size_t OFF_CAB = OFF_CA1 + MATF;            // 128 matrices
static constexpr size_t OFF_X   = OFF_CAB + (size_t)TSTEPS * MATF;  // 2 x 512 ping-pong state
static constexpr size_t OFF_Y   = OFF_X + 2 * NDIM;          // y_obs (512)
static constexpr size_t OFF_YN  = OFF_Y + NDIM;              // y_nat history (128 x 512)
static constexpr size_t OFF_UH  = OFF_YN + (size_t)TSTEPS * NDIM;   // u history (128 x 512)
static constexpr size_t OFF_ES  = OFF_UH + (size_t)TSTEPS * NDIM;   // Esum (512 x 25)
static constexpr size_t OFF_PT  = OFF_ES + (size_t)NDIM * 25;       // cost partials (64)
static constexpr size_t OFF_END = OFF_PT + 64;

// ---------------------------------------------------------------------------
// Zero-init scratch state (state vectors, histories)
// ---------------------------------------------------------------------------
__global__ void zero_kernel(float* __restrict__ p, int n) {
    int i = blockIdx.x * blockDim.x + threadIdx.x;
    if (i < n) p[i] = 0.0f;
}

// ---------------------------------------------------------------------------
// Esum[c, i*5+j] = sum_n E[c, n, i, j]      E: (512, 512, 5, 5) row-major
// ---------------------------------------------------------------------------
__global__ void esum_kernel(const float* __restrict__ E, float* __restrict__ Esum) {
    int gid = blockIdx.x * blockDim.x + threadIdx.x;
    if (gid >= NDIM * 25) return;
    int c = gid / 25, ij = gid % 25;
    const float* base = E + (size_t)c * NDIM * 25 + ij;
    float s = 0.0f;
    for (int n = 0; n < NDIM; ++n) s += base[(size_t)n * 25];
    Esum[gid] = s;
}

// ---------------------------------------------------------------------------
// TDM: async-load a KT x 64 f32 tile (row stride 512 floats) into LDS.
// D# per CDNA5 ISA ch.8: group0 = {count/type/addresses}, group1 = dims/strides.
// 2D tile -> remaining descriptor groups zero.
// ---------------------------------------------------------------------------
__device__ __forceinline__ void tdm_load_tile(const float* gptr, unsigned lds_off) {
    unsigned long long ga = (unsigned long long)(uintptr_t)gptr;
    v4u g0;
    g0.x = 1u;                                        // count=1, is_restore=0, gather off
    g0.y = lds_off;                                   // lds_addr (bytes)
    g0.z = (unsigned)(ga & 0xFFFFFFFFu);              // global_addr[31:0]
    g0.w = (unsigned)((ga >> 32) & 0x1FFFFFFu) | (2u << 30);  // global_addr[56:32] | type=2
    v8i g1;
    g1[0] = 0x20000;                                  // data_size=2 (4B), no multicast/pad
    g1[1] = (int)((NDIM & 0xFFFF) << 16);             // tensor_dim0 lo16 -> bits[63:48]
    g1[2] = (int)((NDIM >> 16) | ((NDIM & 0xFFFF) << 16)); // dim0 hi | tensor_dim1 lo
    g1[3] = (int)((NDIM >> 16) | (64 << 16));         // dim1 hi | tile_dim0 = 64
    g1[4] = KT;                                       // tile_dim1 = KT, tile_dim2 = 0
    g1[5] = NDIM;                                     // tensor_dim0_stride lo32 = 512
    g1[6] = 0;                                        // stride0 hi | stride1 lo
    g1[7] = 0;
    v4i z4 = {0, 0, 0, 0};
    v8i z8 = {0, 0, 0, 0, 0, 0, 0, 0};
    __builtin_amdgcn_tensor_load_to_lds(g0, g1, z4, z4, z8, 0);
}

// ---------------------------------------------------------------------------
// D = CA @ X using V_WMMA_F32_16X16X4_F32 with TDM-staged, double-buffered B.
// grid: (8, 4, 2).  z==0: X=A -> CA_next ; z==1: X=B -> CAB[i]
// block: 256 threads = 8 waves.  Block tile 128(M) x 64(N); each wave: 16x64.
// A frag 16x4 f32: lanes 0-15 hold {K,K+1}, lanes 16-31 hold {K+2,K+3}.
// B frag 4x16 f32: VGPR0 = row K (lanes<16) / K+2 (lanes>=16), VGPR1 = +1.
// ---------------------------------------------------------------------------
__global__ __launch_bounds__(256) void gemm512_wmma(
        const float* __restrict__ CA,
        const float* __restrict__ Amat, const float* __restrict__ Bmat,
        float* __restrict__ outA, float* __restrict__ outB) {
    __shared__ float xs[2][KT * 64];                  // 2 x 8KB double buffer
    const float* __restrict__ X = (blockIdx.z == 0) ? Amat : Bmat;
    float* __restrict__ D       = (blockIdx.z == 0) ? outA : outB;

    const int wave = threadIdx.x >> 5;
    const int lane = threadIdx.x & 31;
    const int l16  = lane & 15;
    const int koff = (lane >> 4) * 2;                 // 0 for lanes 0-15, 2 for 16-31
    const int row0 = blockIdx.y * 128 + wave * 16;
    const int col0 = blockIdx.x * 64;
    constexpr int NSLAB = NDIM / KT;                  // 16

    v8f acc0 = {}; v8f acc1 = {}; v8f acc2 = {}; v8f acc3 = {};

    // wave 0 kicks off the first DMA slab (wave-uniform branch; EXEC ignored by TDM)
    if (threadIdx.x < 32) {
        tdm_load_tile(X + col0, (unsigned)(uintptr_t)&xs[0][0]);
    }

    const float* __restrict__ arow = CA + (size_t)(row0 + l16) * NDIM;
    for (int s = 0; s < NSLAB; ++s) {
        if (threadIdx.x < 32) {
            if (s + 1 < NSLAB) {   // prefetch next slab into the other buffer
                tdm_load_tile(X + (size_t)(s + 1) * KT * NDIM + col0,
                              (unsigned)(uintptr_t)&xs[(s + 1) & 1][0]);
                __builtin_amdgcn_s_wait_tensorcnt((short)1);  // slab s complete
            } else {
                __builtin_amdgcn_s_wait_tensorcnt((short)0);
            }
        }
        __syncthreads();                              // publish slab s to all waves

        const float* __restrict__ xb = &xs[s & 1][0];
        const int kbase = s * KT;
#pragma unroll
        for (int kk = 0; kk < KT; kk += 4) {
            v2f a;
            a.x = arow[kbase + kk + koff];
            a.y = arow[kbase + kk + koff + 1];
            const float* __restrict__ xr0 = xb + (kk + koff) * 64 + l16;
            const float* __restrict__ xr1 = xr0 + 64;
            v2f b0; b0.x = xr0[0];  b0.y = xr1[0];
            v2f b1; b1.x = xr0[16]; b1.y = xr1[16];
            v2f b2; b2.x = xr0[32]; b2.y = xr1[32];
            v2f b3; b3.x = xr0[48]; b3.y = xr1[48];
            acc0 = __builtin_amdgcn_wmma_f32_16x16x4_f32(false, a, false, b0, (short)0, acc0, false, false);
            acc1 = __builtin_amdgcn_wmma_f32_16x16x4_f32(false, a, false, b1, (short)0, acc1, false, false);
            acc2 = __builtin_amdgcn_wmma_f32_16x16x4_f32(false, a, false, b2, (short)0, acc2, false, false);
            acc3 = __builtin_amdgcn_wmma_f32_16x16x4_f32(false, a, false, b3, (short)0, acc3, false, false);
        }
        __syncthreads();                              // all waves done before buffer reuse
    }

    // C/D layout: VGPR r -> M = r (lanes 0-15) or M = 8+r (lanes 16-31); N = lane&15
    const int mbase = row0 + (lane >> 4) * 8;
    v8f accs[4] = {acc0, acc1, acc2, acc3};
#pragma unroll
    for (int j = 0; j < 4; ++j) {
        const int n = col0 + j * 16 + l16;
#pragma unroll
        for (int r = 0; r < 8; ++r) {
            D[(size_t)(mbase + r) * NDIM + n] = accs[j][r];
        }
    }
}

// ---------------------------------------------------------------------------
// x_t = A @ x_{t-1} + B @ u_{t-1} + w_t       (t >= 1)
// 4 rows per block, 64 threads per row, fixed-tree reduction.
// ---------------------------------------------------------------------------
__global__ __launch_bounds__(256) void state_kernel(
        const float* __restrict__ A, const float* __restrict__ B,
        const float* __restrict__ W, const float* __restrict__ xprev,
        const float* __restrict__ uprev, float* __restrict__ xout, int t) {
    __shared__ float red[256];
    const int row  = blockIdx.x * 4 + (threadIdx.x >> 6);
    const int lane = threadIdx.x & 63;
    const float* __restrict__ ar = A + (size_t)row * NDIM;
    const float* __restrict__ br = B + (size_t)row * NDIM;
    float s = 0.0f;
    for (int c = lane; c < NDIM; c += 64) s += ar[c] * xprev[c] + br[c] * uprev[c];
    red[threadIdx.x] = s;
    __syncthreads();
    for (int off = 32; off > 0; off >>= 1) {
        if (lane < off) red[threadIdx.x] += red[threadIdx.x + off];
        __syncthreads();
    }
    if (lane == 0) xout[row] = red[threadIdx.x] + W[(size_t)t * NDIM + row];
}

// ---------------------------------------------------------------------------
// y_obs = C @ x ;  ce[n] = sum_{i=0..t-1} CAB[i][n,:] . u_{t-1-i} ;
// y_nat[t] = y_obs - ce.   float4 (b128) loads — this loop is HBM-bound.
// ---------------------------------------------------------------------------
__global__ __launch_bounds__(256) void ynat_kernel(
        const float* __restrict__ Cm, const float* __restrict__ CAB,
        const float* __restrict__ x, const float* __restrict__ uh,
        float* __restrict__ yobs, float* __restrict__ ynat, int t) {
    __shared__ float red[256];
    const int row  = blockIdx.x * 4 + (threadIdx.x >> 6);
    const int lane = threadIdx.x & 63;
    const int gbase = (threadIdx.x >> 6) << 6;

    const float4* __restrict__ cr = (const float4*)(Cm + (size_t)row * NDIM);
    const float4* __restrict__ x4 = (const float4*)x;
    float sy = 0.0f;
    for (int c = lane; c < NDIM / 4; c += 64) {
        const float4 a = cr[c], b = x4[c];
        sy += a.x * b.x + a.y * b.y + a.z * b.z + a.w * b.w;
    }

    float sce = 0.0f;
    for (int i = 0; i < t; ++i) {                       // i == t term multiplies u_{-1}=0
        const float4* __restrict__ cab = (const float4*)(CAB + (size_t)i * MATF + (size_t)row * NDIM);
        const float4* __restrict__ uv  = (const float4*)(uh + (size_t)(t - 1 - i) * NDIM);
        for (int c = lane; c < NDIM / 4; c += 64) {
            const float4 a = cab[c], b = uv[c];
            sce += a.x * b.x + a.y * b.y + a.z * b.z + a.w * b.w;
        }
    }

    red[threadIdx.x] = sy;
    __syncthreads();
    for (int off = 32; off > 0; off >>= 1) {
        if (lane < off) red[threadIdx.x] += red[threadIdx.x + off];
        __syncthreads();
    }
    const float yo = red[gbase];
    __syncthreads();

    red[threadIdx.x] = sce;
    __syncthreads();
    for (int off = 32; off > 0; off >>= 1) {
        if (lane < off) red[threadIdx.x] += red[threadIdx.x + off];
        __syncthreads();
    }
    if (lane == 0) {
        yobs[row] = yo;
        ynat[(size_t)t * NDIM + row] = yo - red[threadIdx.x];
    }
}

// ---------------------------------------------------------------------------
// u_t[c] = -K[c,:].y_obs + sum_ij Esum[c,ij]*Yfsum[ij] + bias[c]
// Yfsum[i,j] = sum_{MM,mm} F1[i,mm] F2[j,MM] * (MM+mm<5 ? yh_sum[MM+mm] : 0)
// yh_sum[k]  = sum_s y_nat_{t-4+k}[s]
// grid 2 x 256 (one c per thread); yh_sum/Yfsum computed redundantly per block.
// ---------------------------------------------------------------------------
__global__ __launch_bounds__(256) void control_kernel(
        const float* __restrict__ Km, const float* __restrict__ Esum,
        const float* __restrict__ bias,
        const float* __restrict__ sig_m, const float* __restrict__ phi_m,
        const float* __restrict__ sig_M, const float* __restrict__ phi_M,
        const float* __restrict__ yobs, const float* __restrict__ ynat,
        float* __restrict__ uh, int t) {
    __shared__ float red[256];
    __shared__ float yhs[5];
    __shared__ float yf[25];

    for (int k = 0; k < 5; ++k) {
        const int r = t - 4 + k;
        float s = 0.0f;
        if (r >= 0) {
            const float* __restrict__ yr = ynat + (size_t)r * NDIM;
            for (int n = threadIdx.x; n < NDIM; n += 256) s += yr[n];
        }
        red[threadIdx.x] = s;
        __syncthreads();
        for (int off = 128; off > 0; off >>= 1) {
            if (threadIdx.x < off) red[threadIdx.x] += red[threadIdx.x + off];
            __syncthreads();
        }
        if (threadIdx.x == 0) yhs[k] = red[0];
        __syncthreads();
    }

    if (threadIdx.x < 25) {
        const int i = threadIdx.x / 5, j = threadIdx.x % 5;
        const float f1s = powf(sig_m[i], 0.25f);
        float s = 0.0f;
        for (int MM = 0; MM < 5; ++MM) {
            const float f2 = (j == 0) ? 1.0f : powf(sig_M[j - 1], 0.25f) * phi_M[MM * 4 + (j - 1)];
            for (int mm = 0; mm < 5; ++mm) {
                if (MM + mm < 5) {
                    const float f1 = f1s * phi_m[mm * 5 + i];
                    s += f1 * f2 * yhs[MM + mm];
                }
            }
        }
        yf[threadIdx.x] = s;
    }
    __syncthreads();

    const int c = blockIdx.x * 256 + threadIdx.x;
    const float* __restrict__ kr = Km + (size_t)c * NDIM;
    float s = 0.0f;
    for (int n = 0; n < NDIM; ++n) s += kr[n] * yobs[n];
    float up = 0.0f;
    const float* __restrict__ es = Esum + (size_t)c * 25;
    for (int ij = 0; ij < 25; ++ij) up += es[ij] * yf[ij];
    uh[(size_t)t * NDIM + c] = -s + up + bias[c];
}

// ---------------------------------------------------------------------------
// cost_t = y.(Q y) + u.(R u) — two-stage deterministic reduction (no atomics)
// ---------------------------------------------------------------------------
__global__ __launch_bounds__(256) void cost_partial_kernel(
        const float* __restrict__ Q, const float* __restrict__ R,
        const float* __restrict__ yobs, const float* __restrict__ uh,
        float* __restrict__ partial, int t) {
    __shared__ float red[256];
    const float* __restrict__ u = uh + (size_t)t * NDIM;
    float s = 0.0f;
    const int gid = blockIdx.x * 256 + threadIdx.x;
    const int total = 2 * (int)MATF;
    for (int p = gid; p < total; p += 64 * 256) {
        int q = p;
        if (q < (int)MATF) {
            const int n = q >> 9, m = q & 511;
            s += yobs[n] * Q[(size_t)n * NDIM + m] * yobs[m];
        } else {
            q -= (int)MATF;
            const int n = q >> 9, m = q & 511;
            s += u[n] * R[(size_t)n * NDIM + m] * u[m];
        }
    }
    red[threadIdx.x] = s;
    __syncthreads();
    for (int off = 128; off > 0; off >>= 1) {
        if (threadIdx.x < off) red[threadIdx.x] += red[threadIdx.x + off];
        __syncthreads();
    }
    if (threadIdx.x == 0) partial[blockIdx.x] = red[0];
}

__global__ __launch_bounds__(64) void cost_final_kernel(
        const float* __restrict__ partial, float* __restrict__ losses, int t) {
    __shared__ float red[64];
    red[threadIdx.x] = partial[threadIdx.x];
    __syncthreads();
    for (int off = 32; off > 0; off >>= 1) {
        if (threadIdx.x < off) red[threadIdx.x] += red[threadIdx.x + off];
        __syncthreads();
    }
    if (threadIdx.x == 0) losses[t] = red[0];
}

// ---------------------------------------------------------------------------
extern "C" void kernel_launch(void* const* d_in, const int* in_sizes, int n_in,
                              void* d_out, int out_size, void* d_ws, size_t ws_size,
                              hipStream_t stream) {
    (void)in_sizes; (void)n_in; (void)out_size; (void)ws_size;
    const float* A     = (const float*)d_in[0];
    const float* B     = (const float*)d_in[1];
    const float* Cm    = (const float*)d_in[2];
    const float* Q     = (const float*)d_in[3];
    const float* R     = (const float*)d_in[4];
    const float* Km    = (const float*)d_in[5];
    const float* E     = (const float*)d_in[6];
    const float* bias  = (const float*)d_in[7];
    const float* sig_m = (const float*)d_in[8];
    const float* phi_m = (const float*)d_in[9];
    const float* sig_M = (const float*)d_in[10];
    const float* phi_M = (const float*)d_in[11];
    const float* W     = (const float*)d_in[12];
    float* losses = (float*)d_out;

    float* ws   = (float*)d_ws;
    float* ca0  = ws + OFF_CA0;
    float* ca1  = ws + OFF_CA1;
    float* cab  = ws + OFF_CAB;
    float* xbuf = ws + OFF_X;
    float* yobs = ws + OFF_Y;
    float* ynat = ws + OFF_YN;
    float* uh   = ws + OFF_UH;
    float* esum = ws + OFF_ES;
    float* part = ws + OFF_PT;

    // 1) zero state/histories
    {
        const int n = (int)(OFF_END - OFF_X);
        zero_kernel<<<(n + 255) / 256, 256, 0, stream>>>(xbuf, n);
    }
    // 2) Esum precompute
    esum_kernel<<<(NDIM * 25 + 255) / 256, 256, 0, stream>>>(E, esum);

    // 3) CAB chain: CA_0 = C;  CAB[i] = CA_i @ B;  CA_{i+1} = CA_i @ A
    {
        const float* cur = Cm;
        float* bufs[2] = {ca0, ca1};
        for (int i = 0; i < TSTEPS; ++i) {
            float* next = bufs[i & 1];
            gemm512_wmma<<<dim3(8, 4, 2), 256, 0, stream>>>(
                cur, A, B, next, cab + (size_t)i * MATF);
            cur = next;
        }
    }

    // 4) sequential rollout
    for (int t = 0; t < TSTEPS; ++t) {
        float* xcur = xbuf + (size_t)(t & 1) * NDIM;
        if (t > 0) {
            const float* xprev = xbuf + (size_t)((t - 1) & 1) * NDIM;
            const float* uprev = uh + (size_t)(t - 1) * NDIM;
            state_kernel<<<NDIM / 4, 256, 0, stream>>>(A, B, W, xprev, uprev, xcur, t);
        }
        ynat_kernel<<<NDIM / 4, 256, 0, stream>>>(Cm, cab, xcur, uh, yobs, ynat, t);
        control_kernel<<<2, 256, 0, stream>>>(Km, esum, bias, sig_m, phi_m, sig_M, phi_M,
                                              yobs, ynat, uh, t);
        cost_partial_kernel<<<64, 256, 0, stream>>>(Q, R, yobs, uh, part, t);
        cost_final_kernel<<<1, 64, 0, stream>>>(part, losses, t);
    }
}